// Pointer_9302899163660
// MI455X (gfx1250) — compile-verified
//
#include <hip/hip_runtime.h>
#include <hip/hip_bf16.h>

typedef __attribute__((ext_vector_type(2))) float v2f;
typedef __attribute__((ext_vector_type(8))) float v8f;

#define H 768
#define BB 2
#define LL 32
#define MM 64
#define NEGV -1000000.0f

// ---- gfx1250 hardware tanh (TRANS op) ---------------------------------------
__device__ __forceinline__ float hw_tanh(float x) {
#if __has_builtin(__builtin_amdgcn_tanh_f32)
    return __builtin_amdgcn_tanh_f32(x);
#elif __has_builtin(__builtin_amdgcn_tanhf)
    return __builtin_amdgcn_tanhf(x);
#else
    float r;
    // TRANS RAW hazard: 1 independent op required before result use (ISA 7.4.3)
    asm volatile("v_tanh_f32 %0, %1\n\tv_nop\n\tv_nop" : "=v"(r) : "v"(x));
    return r;
#endif
}

// ---- async global->LDS b128 (ASYNCcnt-tracked) ------------------------------
__device__ __forceinline__ void async_ld_b128(void* lds_dst, const void* gsrc) {
    // Low 32 bits of a generic pointer to LDS == wave-relative LDS byte offset.
    unsigned      ldsoff = (unsigned)(uintptr_t)lds_dst;
    unsigned long long ga = (unsigned long long)(uintptr_t)gsrc;
    asm volatile("global_load_async_to_lds_b128 %0, %1, off"
                 :: "v"(ldsoff), "v"(ga) : "memory");
}
__device__ __forceinline__ void wait_async0() {
    asm volatile("s_wait_asynccnt 0x0" ::: "memory");
}

// out[R, H] = A[R, H] @ W[H, H]^T + bias   (torch Linear semantics)
// Each wave produces a 16x32 output tile: one A fragment feeds two independent
// V_WMMA_F32_16X16X4_F32 ops (2 accumulators) -> fewer loads per WMMA + ILP.
// A-tile (32-bit 16x4): m = lane%16, kbase = h0 + 2*(lane/16), a = A[m][kbase..+1].
// B-tile (4x16): n = lane%16, b[v] = W[c0+n][kbase+v] (contiguous float2).
// C/D layout: D[v + 8*(lane/16)][lane%16].
__global__ void __launch_bounds__(256)
proj_wmma_kernel(const float* __restrict__ A,
                 const float* __restrict__ W,
                 const float* __restrict__ bias,
                 float* __restrict__ out)
{
    const int lane = threadIdx.x & 31;
    const int half = lane >> 4;         // 0: lanes 0-15, 1: lanes 16-31
    const int lmod = lane & 15;
    const int wave = threadIdx.x >> 5;  // 8 waves per block
    const int tile = blockIdx.x * 8 + wave;

    const int colTiles = H / 32;        // 24 (32-wide N tiles)
    const int rt = tile / colTiles;
    const int ct = tile % colTiles;
    const int r0 = rt * 16;
    const int c0 = ct * 32;

    const float* __restrict__ arow  = A + (size_t)(r0 + lmod) * H;
    const float* __restrict__ wrow0 = W + (size_t)(c0 + lmod) * H;
    const float* __restrict__ wrow1 = W + (size_t)(c0 + 16 + lmod) * H;

    v8f acc0 = {};
    v8f acc1 = {};
    #pragma unroll 4
    for (int h0 = 0; h0 < H; h0 += 4) {
        const int kb = h0 + 2 * half;
        v2f a  = *(const v2f*)(arow  + kb);
        v2f b0 = *(const v2f*)(wrow0 + kb);
        v2f b1 = *(const v2f*)(wrow1 + kb);
        // (neg_a, A, neg_b, B, c_mod, C, reuse_a, reuse_b)
        acc0 = __builtin_amdgcn_wmma_f32_16x16x4_f32(
            false, a, false, b0, (short)0, acc0, false, false);
        acc1 = __builtin_amdgcn_wmma_f32_16x16x4_f32(
            false, a, false, b1, (short)0, acc1, false, false);
    }

    const float bv0 = bias[c0 + lmod];
    const float bv1 = bias[c0 + 16 + lmod];
    #pragma unroll
    for (int v = 0; v < 8; ++v) {
        const int row = r0 + v + 8 * half;
        out[(size_t)row * H + c0 + lmod]      = acc0[v] + bv0;
        out[(size_t)row * H + c0 + 16 + lmod] = acc1[v] + bv1;
    }
}

// One block per (b,l,k). h-row and vw staged to LDS via async b128 copies;
// each of the 8 waves covers 8 of the 64 memory slots; hardware v_tanh_f32
// does the transcendental; 32-lane shuffle reduction per slot.
__global__ void __launch_bounds__(256)
attn_tanh_kernel(const float* __restrict__ hproj,   // [B*L*L, H]
                 const float* __restrict__ mproj,   // [B*M, H]
                 const int*   __restrict__ ce_mask, // [B, M]
                 const float* __restrict__ vw,      // [H]
                 const float* __restrict__ vbp,     // [1]
                 float*       __restrict__ out)     // [B*L*L, M]
{
    __shared__ float h_s[H];
    __shared__ float v_s[H];

    const int blk = blockIdx.x;          // (b*L + l)*L + k
    const int b   = blk / (LL * LL);

    const float* __restrict__ hrow = hproj + (size_t)blk * H;
    {
        const int i = threadIdx.x * 4;   // 192 lanes move 16B each per array
        if (i < H) {
            async_ld_b128(&h_s[i], hrow + i);
            async_ld_b128(&v_s[i], vw + i);
        }
    }
    wait_async0();
    __syncthreads();

    const int wave = threadIdx.x >> 5;
    const int lane = threadIdx.x & 31;
    const float vb = vbp[0];

    #pragma unroll
    for (int j = 0; j < 8; ++j) {
        const int m = wave + 8 * j;      // 0..63
        const float* __restrict__ mrow = mproj + (size_t)(b * MM + m) * H;

        float acc = 0.0f;
        #pragma unroll 4
        for (int h = lane; h < H; h += 32)
            acc += hw_tanh(h_s[h] + mrow[h]) * v_s[h];

        // wave32 reduction
        #pragma unroll
        for (int off = 16; off > 0; off >>= 1)
            acc += __shfl_xor(acc, off, 32);

        if (lane == 0) {
            const float r = ce_mask[b * MM + m] ? (acc + vb) : NEGV;
            out[(size_t)blk * MM + m] = r;
        }
    }
}

extern "C" void kernel_launch(void* const* d_in, const int* in_sizes, int n_in,
                              void* d_out, int out_size, void* d_ws, size_t ws_size,
                              hipStream_t stream)
{
    const float* hs      = (const float*)d_in[0]; // [B,L,L,H]
    const float* memory  = (const float*)d_in[1]; // [B,M,H]
    const int*   ce_mask = (const int*)  d_in[2]; // [B,M]
    const float* Wh      = (const float*)d_in[3]; // [H,H]
    const float* bh      = (const float*)d_in[4]; // [H]
    const float* Wm      = (const float*)d_in[5]; // [H,H]
    const float* bm      = (const float*)d_in[6]; // [H]
    const float* vw      = (const float*)d_in[7]; // [1,H]
    const float* vb      = (const float*)d_in[8]; // [1]
    float*       out     = (float*)d_out;         // [B,L,L,M]

    const int Rh = BB * LL * LL;  // 2048 rows for hs projection
    const int Rm = BB * MM;       // 128 rows for memory projection

    float* hproj = (float*)d_ws;               // 2048*768*4 = 6.29 MB
    float* mproj = hproj + (size_t)Rh * H;     // +128*768*4 = 0.39 MB

    // hproj = hs @ Wh^T + bh : 128 row-tiles * 24 (16x32) col-tiles = 3072 wave-tiles
    {
        const int tiles = (Rh / 16) * (H / 32);
        proj_wmma_kernel<<<tiles / 8, 256, 0, stream>>>(hs, Wh, bh, hproj);
    }
    // mproj = memory @ Wm^T + bm : 8 * 24 = 192 wave-tiles
    {
        const int tiles = (Rm / 16) * (H / 32);
        proj_wmma_kernel<<<tiles / 8, 256, 0, stream>>>(memory, Wm, bm, mproj);
    }
    // fused hw-tanh + v-dot + mask
    attn_tanh_kernel<<<Rh, 256, 0, stream>>>(hproj, mproj, ce_mask, vw, vb, out);
}